// LaplacianPyramidplan3_46986942218805
// MI455X (gfx1250) — compile-verified
//
#include <hip/hip_runtime.h>
#include <hip/hip_bf16.h>

// ---------------- problem constants ----------------
#define IMG_H   256
#define IMG_W   256
#define CH      64
#define BATCH   16
#define NPIX    65536            // IMG_H*IMG_W
#define NROWS   1024             // BATCH*CH
#define KSPLIT  64               // split-K factor
#define KCHUNK  1024             // NPIX / KSPLIT
#define KSTEP   64               // LDS tile width
#define LDSW    68               // padded LDS row stride (dwords); TDM pad: 64 dwords + 4 pad
#define NSTEPS  (KCHUNK / KSTEP) // 16

typedef __attribute__((ext_vector_type(2))) float v2f;
typedef __attribute__((ext_vector_type(8))) float v8f;
typedef __attribute__((ext_vector_type(4))) unsigned int u32x4;
typedef __attribute__((ext_vector_type(8))) unsigned int u32x8;

// ---------------- compile-time composite 13-tap kernel ----------------
struct Taps { float w[13]; };
__host__ __device__ constexpr Taps make_taps() {
    const float g3[3] = {0.310975f, 0.378049f, 0.310975f};
    const float g5[5] = {0.153168f, 0.221553f, 0.250558f, 0.221553f, 0.153168f};
    const float g7[7] = {0.093687f, 0.138036f, 0.174172f, 0.188209f,
                         0.174172f, 0.138036f, 0.093687f};
    float t7[7] = {};
    for (int i = 0; i < 3; ++i)
        for (int j = 0; j < 5; ++j) t7[i + j] += g3[i] * g5[j];
    Taps t{};
    for (int i = 0; i < 7; ++i)
        for (int j = 0; j < 7; ++j) t.w[i + j] += t7[i] * g7[j];
    return t;
}
constexpr Taps TAPS = make_taps();

// ---------------- kernel 1: context = x - blur13(x) ----------------
__global__ __launch_bounds__(256) void k_pyramid_context(
    const float* __restrict__ x, float* __restrict__ ctx) {
    constexpr int R = 6, TILE = 32, IN = 44;
    __shared__ float smem_in[IN][IN + 1];
    __shared__ float hbuf[IN][TILE + 1];

    const int blk = blockIdx.x;
    const int tx = blk & 7, ty = (blk >> 3) & 7, bc = blk >> 6;
    const int gx0 = tx * TILE - R, gy0 = ty * TILE - R;
    const float* src = x + (size_t)bc * NPIX;

    for (int idx = threadIdx.x; idx < IN * IN; idx += 256) {
        const int r = idx / IN, c = idx % IN;
        const int gy = gy0 + r, gx = gx0 + c;
        float v = 0.0f;
        if ((unsigned)gy < IMG_H && (unsigned)gx < IMG_W) v = src[gy * IMG_W + gx];
        smem_in[r][c] = v;
    }
    __syncthreads();

    for (int idx = threadIdx.x; idx < IN * TILE; idx += 256) {
        const int r = idx / TILE, c = idx % TILE;
        float s = 0.0f;
#pragma unroll
        for (int t = 0; t < 13; ++t) s += TAPS.w[t] * smem_in[r][c + t];
        hbuf[r][c] = s;
    }
    __syncthreads();

    float* dst = ctx + (size_t)bc * NPIX;
    for (int idx = threadIdx.x; idx < TILE * TILE; idx += 256) {
        const int r = idx / TILE, c = idx % TILE;
        float s = 0.0f;
#pragma unroll
        for (int t = 0; t < 13; ++t) s += TAPS.w[t] * hbuf[r + t][c];
        dst[(size_t)(gy0 + R + r) * IMG_W + (gx0 + R + c)] = smem_in[r + R][c + R] - s;
    }
}

// ---------------- kernel 2: per-row softmax stats --------------------------
__global__ __launch_bounds__(256) void k_softmax_stats(
    const float* __restrict__ ctx, float* __restrict__ rowmax,
    float* __restrict__ rowsum) {
    __shared__ float red[256];
    const int row = blockIdx.x;
    const float* p = ctx + (size_t)row * NPIX;

    float m = -3.402823466e38f;
    for (int i = threadIdx.x; i < NPIX; i += 256) m = fmaxf(m, p[i]);
    red[threadIdx.x] = m;
    __syncthreads();
    for (int s = 128; s > 0; s >>= 1) {
        if (threadIdx.x < s) red[threadIdx.x] = fmaxf(red[threadIdx.x], red[threadIdx.x + s]);
        __syncthreads();
    }
    m = red[0];
    __syncthreads();

    float s = 0.0f;
    for (int i = threadIdx.x; i < NPIX; i += 256) s += __expf(p[i] - m);
    red[threadIdx.x] = s;
    __syncthreads();
    for (int st = 128; st > 0; st >>= 1) {
        if (threadIdx.x < st) red[threadIdx.x] += red[threadIdx.x + st];
        __syncthreads();
    }
    if (threadIdx.x == 0) { rowmax[row] = m; rowsum[row] = red[0]; }
}

// ---------------- TDM helper: issue one 64x64 f32 tile load into LDS -------
// D# per CDNA5 ISA §8 (CDNA5 ISA):
//  group0: [1:0]=count=1, [63:32]=lds_addr, [120:64]=global_addr, [127:126]=type=2
//  group1: data_size=4B (2), pad_enable, pad_interval=64dw (5), pad_amount=4dw (3),
//          tensor_dim0=65536, tensor_dim1=64, tile 64x64, dim0_stride=65536
// 2-D tensor -> 2-operand form (VADDR2/3 = NULL).
__device__ __forceinline__ void tdm_issue_tile(unsigned long long gaddr,
                                               unsigned lds_addr,
                                               const u32x8& g1) {
    const u32x4 g0 = {1u,                                   // count=1 (valid, load)
                      lds_addr,                              // LDS byte address
                      (unsigned)gaddr,                       // global addr [31:0]
                      ((unsigned)(gaddr >> 32) & 0x1FFFFFFu) // global addr [56:32]
                          | (2u << 30)};                     // type=2 ("image")
    asm volatile("tensor_load_to_lds %0, %1" ::"s"(g0), "s"(g1) : "memory");
}

// ---------------- kernel 3: split-K f32 WMMA GEMM, TDM double-buffered -----
__global__ __launch_bounds__(256) void k_attn_gemm_splitk(
    const float* __restrict__ ctx, const float* __restrict__ rowmax,
    float* __restrict__ partial) {
    __shared__ float tile[2][64 * LDSW];

    const int b  = blockIdx.x / KSPLIT;
    const int ks = blockIdx.x % KSPLIT;
    const float* cb = ctx + (size_t)b * CH * NPIX + (size_t)ks * KCHUNK;

    const int tid   = threadIdx.x;
    const int w     = tid >> 5;
    const int lane  = tid & 31;
    const int lid   = lane & 15;
    const int khalf = lane >> 4;
    const int d0    = (w & 3) * 16;
    const int e0    = (w >> 2) * 32;

    const float rm = rowmax[b * CH + d0 + lid];
    v8f acc0 = {}; v8f acc1 = {};

    // constant descriptor group 1
    const u32x8 g1 = {
        (2u << 16) | (1u << 20) | (5u << 22) | (3u << 25), // 4B elems, pad 4dw per 64dw
        0u,                                                 // atomic_bar=0, tdim0 lo16=0
        0x00400001u,                                        // tdim0 hi16=1 (65536), tdim1 lo16=64
        0x00400000u,                                        // tdim1 hi16=0, tile_dim0=64
        64u,                                                // tile_dim1=64, tile_dim2=0
        (unsigned)NPIX,                                     // tensor_dim0_stride lo32 = 65536
        0u, 0u};

    const unsigned long long gbase = (unsigned long long)(uintptr_t)cb;
    const unsigned lds0 =
        (unsigned)(size_t)(__attribute__((address_space(3))) char*)&tile[0][0];
    const unsigned lds1 =
        (unsigned)(size_t)(__attribute__((address_space(3))) char*)&tile[1][0];

    // prologue: DMA tile 0
    if (tid < 32) tdm_issue_tile(gbase, lds0, g1);

    for (int s = 0; s < NSTEPS; ++s) {
        if (tid < 32) {
            if (s + 1 < NSTEPS) {
                // prefetch next tile into the other buffer, then wait for tile s
                tdm_issue_tile(gbase + (unsigned long long)(s + 1) * (KSTEP * 4),
                               ((s + 1) & 1) ? lds1 : lds0, g1);
                __builtin_amdgcn_s_wait_tensorcnt(1);
            } else {
                __builtin_amdgcn_s_wait_tensorcnt(0);
            }
        }
        __syncthreads();  // tile[s&1] ready for all waves

        const float* t = &tile[s & 1][0];
#pragma unroll
        for (int kk = 0; kk < KSTEP; kk += 4) {
            const int col = kk + khalf * 2;
            v2f a = *(const v2f*)&t[(d0 + lid) * LDSW + col];
            a.x = __expf(a.x - rm);
            a.y = __expf(a.y - rm);
            const v2f b0 = *(const v2f*)&t[(e0 + lid) * LDSW + col];
            const v2f b1 = *(const v2f*)&t[(e0 + 16 + lid) * LDSW + col];
            acc0 = __builtin_amdgcn_wmma_f32_16x16x4_f32(
                false, a, false, b0, (short)0, acc0, false, false);
            acc1 = __builtin_amdgcn_wmma_f32_16x16x4_f32(
                false, a, false, b1, (short)0, acc1, false, false);
        }
        __syncthreads();  // done reading tile[s&1] before it is DMA-overwritten
    }

    float* pb = partial + (size_t)blockIdx.x * 4096;
#pragma unroll
    for (int i = 0; i < 8; ++i) {
        const int d = d0 + i + khalf * 8;
        pb[d * 64 + (e0 + lid)]      = acc0[i];
        pb[d * 64 + (e0 + 16 + lid)] = acc1[i];
    }
}

// ---------------- kernel 4: split-K reduce + 1/rowsum scale ----------------
__global__ __launch_bounds__(256) void k_splitk_reduce(
    const float* __restrict__ partial, const float* __restrict__ rowsum,
    float* __restrict__ out) {
    const int g = blockIdx.x * 256 + threadIdx.x;
    const int b = g >> 12;
    const int d = (g >> 6) & 63;
    const float* p = partial + (size_t)b * KSPLIT * 4096 + (g & 4095);
    float s = 0.0f;
#pragma unroll 8
    for (int k = 0; k < KSPLIT; ++k) s += p[(size_t)k * 4096];
    out[g] = s / rowsum[b * CH + d];
}

// ---------------- launcher ----------------
extern "C" void kernel_launch(void* const* d_in, const int* in_sizes, int n_in,
                              void* d_out, int out_size, void* d_ws, size_t ws_size,
                              hipStream_t stream) {
    const float* x   = (const float*)d_in[0];
    float*       out = (float*)d_out;

    float* ctx     = (float*)d_ws;
    float* rowmax  = ctx + (size_t)NROWS * NPIX;
    float* rowsum  = rowmax + NROWS;
    float* partial = rowsum + NROWS;

    k_pyramid_context<<<NROWS * 64, 256, 0, stream>>>(x, ctx);
    k_softmax_stats<<<NROWS, 256, 0, stream>>>(ctx, rowmax, rowsum);
    k_attn_gemm_splitk<<<BATCH * KSPLIT, 256, 0, stream>>>(ctx, rowmax, partial);
    k_splitk_reduce<<<(BATCH * CH * CH) / 256, 256, 0, stream>>>(partial, rowsum, out);
}